// BertSelfAttention_18485539242538
// MI455X (gfx1250) — compile-verified
//
#include <hip/hip_runtime.h>
#include <math.h>

typedef float v2f __attribute__((ext_vector_type(2)));
typedef float v8f __attribute__((ext_vector_type(8)));

// D = A(16x4) * B(4x16) + C, fp32, wave32.
#define WMMA_F32(a, b, c) \
  __builtin_amdgcn_wmma_f32_16x16x4_f32(false, (a), false, (b), (short)0, (c), false, false)

// Async copy: one 16-byte chunk per lane, memory -> LDS, tracked by ASYNCcnt.
#define ASYNC_B128(lds_u32, gaddr_u64)                                  \
  asm volatile("global_load_async_to_lds_b128 %0, %1, off"              \
               :: "v"(lds_u32), "v"(gaddr_u64) : "memory")
#define WAIT_ASYNC_16() asm volatile("s_wait_asynccnt 0x10" ::: "memory")
#define WAIT_ASYNC_0()  asm volatile("s_wait_asynccnt 0x0"  ::: "memory")

// B_ENC=8, BEAM=4, H=16, S_ENC=4096, S_DEC=1020, Q=4, D=64, HID=1024
// dec keys per beam = 1020 cached + 4 new = 1024 = 64 tiles of 16.
#define LDS_STRIDE 68   // padded row stride (floats): bank = (4*row + col) % 64

// ---------------------------------------------------------------------------
// Kernel A: QKV projection.  out = hidden @ W^T + bias, M=128, N=K=1024.
// One wave computes one 16x16 output tile with 256 chained f32 WMMAs.
// ---------------------------------------------------------------------------
__global__ __launch_bounds__(32) void qkv_proj_kernel(
    const float* __restrict__ hidden,
    const float* __restrict__ Wq, const float* __restrict__ bq,
    const float* __restrict__ Wk, const float* __restrict__ bk,
    const float* __restrict__ Wv, const float* __restrict__ bv,
    float* __restrict__ q_ws, float* __restrict__ k_ws, float* __restrict__ v_ws)
{
    const int lane  = threadIdx.x & 31;
    const int mb    = blockIdx.x * 16;
    const int nb    = blockIdx.y * 16;
    const int which = blockIdx.z;        // 0=Q,1=K,2=V

    const float* W    = (which == 0) ? Wq : (which == 1) ? Wk : Wv;
    const float* bias = (which == 0) ? bq : (which == 1) ? bk : bv;

    const int lm   = lane & 15;
    const int hi   = lane >> 4;
    const int koff = hi << 1;

    const float* aptr = hidden + (mb + lm) * 1024 + koff;
    const float* bptr = W      + (nb + lm) * 1024 + koff;

    v8f acc = {0.f, 0.f, 0.f, 0.f, 0.f, 0.f, 0.f, 0.f};
    #pragma unroll 8
    for (int k = 0; k < 1024; k += 4) {
        v2f a = *(const v2f*)(aptr + k);
        v2f b = *(const v2f*)(bptr + k);
        acc = WMMA_F32(a, b, acc);
    }

    const float bval = bias[nb + lm];
    const int col = nb + lm;
    const int h = col >> 6, d = col & 63;
    #pragma unroll
    for (int r = 0; r < 8; ++r) {
        float val = acc[r] + bval;
        int row = mb + r + 8 * hi;          // 0..127 = b*4 + t
        int b = row >> 2, t = row & 3;
        if (which == 0) {
            int benc = b >> 2, beam = b & 3;
            q_ws[((benc * 16 + h) * 16 + (beam * 4 + t)) * 64 + d] = val;
        } else {
            float* dst = (which == 1) ? k_ws : v_ws;
            dst[((b * 16 + h) * 4 + t) * 64 + d] = val;
        }
    }
}

// ---------------------------------------------------------------------------
// Kernel B: fused attention, one workgroup per (b_enc, h), 4 waves.
// Wave w: enc keys [w*1024, w*1024+1024) for all 16 rows, plus beam w's
// decoder cache (other beams' rows masked).  128-tile software pipeline:
// async-copy tile t+1 (K+V, 16 x b128) into LDS while consuming tile t.
// ---------------------------------------------------------------------------
__global__ __launch_bounds__(128) void attn_kernel(
    const float* __restrict__ q_ws, const float* __restrict__ k_ws,
    const float* __restrict__ v_ws,
    const float* __restrict__ enc_k, const float* __restrict__ enc_v,
    const float* __restrict__ dec_k, const float* __restrict__ dec_v,
    const float* __restrict__ mask, float* __restrict__ out)
{
    const int benc = blockIdx.x;       // 0..7
    const int h    = blockIdx.y;       // 0..15
    const int tid  = threadIdx.x;
    const int w    = tid >> 5;         // wave id == beam id for dec part
    const int lane = tid & 31;
    const int lm   = lane & 15;
    const int hi   = lane >> 4;
    const int koff = hi << 1;
    const float scale = 0.125f;        // 1/sqrt(64)

    __shared__ float sK[4][2][16 * LDS_STRIDE];   // double-buffered K tiles
    __shared__ float sV[4][2][16 * LDS_STRIDE];   // double-buffered V tiles
    __shared__ float sP[4][16 * 20];              // P relayout, stride 20
    __shared__ float smx[4][16];
    __shared__ float slx[4][16];
    __shared__ float sctx[4][16][64];

    // Preload Q tile (16x64) as 16 A-operands (K=4 each).
    const float* qbase = q_ws + (benc * 16 + h) * 16 * 64;
    v2f aQ[16];
    #pragma unroll
    for (int i = 0; i < 16; ++i)
        aQ[i] = *(const v2f*)(qbase + lm * 64 + 4 * i + koff);

    const float* Kb = enc_k + (size_t)(benc * 16 + h) * 4096 * 64;
    const float* Vb = enc_v + (size_t)(benc * 16 + h) * 4096 * 64;
    const int bglob = benc * 4 + w;
    const float* Kd = dec_k + (size_t)(bglob * 16 + h) * 1020 * 64;
    const float* Vd = dec_v + (size_t)(bglob * 16 + h) * 1020 * 64;
    const float* Kn = k_ws + (bglob * 16 + h) * 4 * 64;
    const float* Vn = v_ws + (bglob * 16 + h) * 4 * 64;

    // global row pointers for tile t (0..63 enc, 64..127 dec incl. new keys)
    auto kRow = [&](int t, int r) -> const float* {
        if (t < 64) return Kb + (size_t)(w * 1024 + t * 16 + r) * 64;
        int ks = (t - 64) * 16 + r;
        return (ks < 1020) ? Kd + (size_t)ks * 64 : Kn + (size_t)(ks - 1020) * 64;
    };
    auto vRow = [&](int t, int r) -> const float* {
        if (t < 64) return Vb + (size_t)(w * 1024 + t * 16 + r) * 64;
        int ks = (t - 64) * 16 + r;
        return (ks < 1020) ? Vd + (size_t)ks * 64 : Vn + (size_t)(ks - 1020) * 64;
    };

    // issue async copies for one K tile + one V tile (8 + 8 x b128)
    auto stage = [&](int t) {
        const int p = t & 1;
        const uint32_t kb = (uint32_t)(uintptr_t)&sK[w][p][0];
        const uint32_t vb = (uint32_t)(uintptr_t)&sV[w][p][0];
        #pragma unroll
        for (int i = 0; i < 8; ++i) {
            const int row = 2 * i + hi;
            const uint32_t loff = (uint32_t)((row * LDS_STRIDE + lm * 4) * 4);
            ASYNC_B128(kb + loff,
                       (unsigned long long)(uintptr_t)(kRow(t, row) + lm * 4));
            ASYNC_B128(vb + loff,
                       (unsigned long long)(uintptr_t)(vRow(t, row) + lm * 4));
        }
    };

    float m8[8], l8[8];
    v8f ctx[4];
    #pragma unroll
    for (int r = 0; r < 8; ++r) { m8[r] = -1e30f; l8[r] = 0.f; }
    #pragma unroll
    for (int tc = 0; tc < 4; ++tc) ctx[tc] = (v8f){0,0,0,0,0,0,0,0};

    // attention_mask row pointers, one per C-layout register row
    const float* mrow[8];
    #pragma unroll
    for (int r = 0; r < 8; ++r) {
        int row = r + 8 * hi;
        int bb  = benc * 4 + (row >> 2);
        mrow[r] = mask + (bb * 4 + (row & 3)) * 5120;
    }

    v2f aP[4];
    auto online = [&](v8f& s) {
        float corr[8];
        #pragma unroll
        for (int r = 0; r < 8; ++r) {
            float t = s[r];
            t = fmaxf(t, __shfl_xor(t, 1, 32));
            t = fmaxf(t, __shfl_xor(t, 2, 32));
            t = fmaxf(t, __shfl_xor(t, 4, 32));
            t = fmaxf(t, __shfl_xor(t, 8, 32));        // row max (16-lane half)
            float mnew = fmaxf(m8[r], t);
            corr[r] = __expf(m8[r] - mnew);
            float p = __expf(s[r] - mnew);
            s[r] = p;
            float su = p;
            su += __shfl_xor(su, 1, 32);
            su += __shfl_xor(su, 2, 32);
            su += __shfl_xor(su, 4, 32);
            su += __shfl_xor(su, 8, 32);
            l8[r] = l8[r] * corr[r] + su;
            m8[r] = mnew;
        }
        #pragma unroll
        for (int tc = 0; tc < 4; ++tc)
            #pragma unroll
            for (int r = 0; r < 8; ++r) ctx[tc][r] *= corr[r];
        #pragma unroll
        for (int r = 0; r < 8; ++r) sP[w][(r + 8 * hi) * 20 + lm] = s[r];
        __builtin_amdgcn_wave_barrier();
        #pragma unroll
        for (int j = 0; j < 4; ++j)
            aP[j] = *(const v2f*)(&sP[w][lm * 20 + 4 * j + koff]);
    };

    // ---- 128-tile pipeline: 64 enc tiles then 64 dec tiles
    stage(0);
    for (int t = 0; t < 128; ++t) {
        if (t < 127) { stage(t + 1); WAIT_ASYNC_16(); }
        else         { WAIT_ASYNC_0(); }

        const float* kT = &sK[w][t & 1][0];
        const float* vT = &sV[w][t & 1][0];

        v8f s = {0,0,0,0,0,0,0,0};
        #pragma unroll
        for (int i = 0; i < 16; ++i) {
            v2f b = *(const v2f*)(kT + lm * LDS_STRIDE + 4 * i + koff);
            s = WMMA_F32(aQ[i], b, s);
        }

        if (t < 64) {
            const int kv = w * 1024 + t * 16 + lm;
            #pragma unroll
            for (int r = 0; r < 8; ++r) s[r] = s[r] * scale + mrow[r][kv];
        } else {
            const int kv = 4096 + (t - 64) * 16 + lm;
            #pragma unroll
            for (int r = 0; r < 8; ++r) {
                int row = r + 8 * hi;
                s[r] = ((row >> 2) != w) ? -1e30f : s[r] * scale + mrow[r][kv];
            }
        }

        online(s);

        #pragma unroll
        for (int tc = 0; tc < 4; ++tc) {
            #pragma unroll
            for (int j = 0; j < 4; ++j) {
                const int rr = 4 * j + koff;
                v2f b;
                b.x = vT[rr * LDS_STRIDE + 16 * tc + lm];
                b.y = vT[(rr + 1) * LDS_STRIDE + 16 * tc + lm];
                ctx[tc] = WMMA_F32(aP[j], b, ctx[tc]);
            }
        }
    }

    // ---- merge the 4 per-wave partials
    if (lm == 0) {
        #pragma unroll
        for (int r = 0; r < 8; ++r) {
            int row = r + 8 * hi;
            smx[w][row] = m8[r];
            slx[w][row] = l8[r];
        }
    }
    #pragma unroll
    for (int tc = 0; tc < 4; ++tc)
        #pragma unroll
        for (int r = 0; r < 8; ++r)
            sctx[w][r + 8 * hi][16 * tc + lm] = ctx[tc][r];
    __syncthreads();

    // 16 rows x 64 dims = 1024 outputs over 128 threads
    #pragma unroll
    for (int e = 0; e < 8; ++e) {
        int flat = tid + e * 128;
        int row = flat >> 6, d = flat & 63;
        float mx = fmaxf(fmaxf(smx[0][row], smx[1][row]),
                         fmaxf(smx[2][row], smx[3][row]));
        float lsum = 0.f, num = 0.f;
        #pragma unroll
        for (int ww = 0; ww < 4; ++ww) {
            float f = __expf(smx[ww][row] - mx);
            lsum += slx[ww][row] * f;
            num  += sctx[ww][row][d] * f;
        }
        int beam = row >> 2, t = row & 3;
        out[(((benc * 4 + beam) * 4 + t) * 16 + h) * 64 + d] = num / lsum;
    }
}

// ---------------------------------------------------------------------------
extern "C" void kernel_launch(void* const* d_in, const int* in_sizes, int n_in,
                              void* d_out, int out_size, void* d_ws, size_t ws_size,
                              hipStream_t stream) {
    (void)in_sizes; (void)n_in; (void)out_size; (void)ws_size;
    const float* hidden = (const float*)d_in[0];
    const float* mask   = (const float*)d_in[1];
    const float* enc_k  = (const float*)d_in[2];
    const float* enc_v  = (const float*)d_in[3];
    const float* dec_k  = (const float*)d_in[4];
    const float* dec_v  = (const float*)d_in[5];
    const float* Wq     = (const float*)d_in[6];
    const float* bq     = (const float*)d_in[7];
    const float* Wk     = (const float*)d_in[8];
    const float* bk     = (const float*)d_in[9];
    const float* Wv     = (const float*)d_in[10];
    const float* bv     = (const float*)d_in[11];
    float* out = (float*)d_out;
    float* ws  = (float*)d_ws;
    float* q_ws = ws;             // 8*16*16*64  = 131072 floats
    float* k_ws = ws + 131072;    // 32*16*4*64  = 131072 floats
    float* v_ws = ws + 262144;    // 32*16*4*64  = 131072 floats

    qkv_proj_kernel<<<dim3(8, 64, 3), 32, 0, stream>>>(
        hidden, Wq, bq, Wk, bk, Wv, bv, q_ws, k_ws, v_ws);
    attn_kernel<<<dim3(8, 16), 128, 0, stream>>>(
        q_ws, k_ws, v_ws, enc_k, enc_v, dec_k, dec_v, mask, out);
}